// SpatialDerivativeOperator_16939351015514
// MI455X (gfx1250) — compile-verified
//
#include <hip/hip_runtime.h>
#include <hip/hip_bf16.h>

// ---------------------------------------------------------------------------
// SpatialDerivativeOperator for MI455X (gfx1250).
//
// Memory-bound sparse gather/scatter (no dense matmul -> WMMA inapplicable).
// Strategy:
//   * Stream edge data (768 MB) with non-temporal 128-bit loads so it does
//     not evict the L2-resident x table (16 MB) / accumulator (8 MB) from the
//     192 MB global L2.
//   * Accumulate (sum, count) as an interleaved float2 per node with native
//     no-return global_atomic_add_f32 (unsafeAtomicAdd) - both atomics of an
//     edge land on the same 8 bytes.
//   * Finalize: out[i] = cnt>0 ? sum/cnt : 0.
// ---------------------------------------------------------------------------

typedef __attribute__((ext_vector_type(4))) float v4f;
typedef __attribute__((ext_vector_type(2))) float v2f;
typedef __attribute__((ext_vector_type(2))) unsigned long long v2u64;

__global__ void sdo_zero_kernel(float* __restrict__ acc, long long n_floats) {
    long long i = ((long long)blockIdx.x * blockDim.x + threadIdx.x) * 4;
    long long stride = (long long)gridDim.x * blockDim.x * 4;
    for (; i + 3 < n_floats; i += stride) {
        v4f z = (v4f)(0.0f);
        *(v4f*)(acc + i) = z;
    }
}

__device__ __forceinline__ void sdo_process_edge(const float* __restrict__ x,
                                                 float* __restrict__ acc,
                                                 long long s, long long d,
                                                 float a) {
    // x has row stride 4; we need column 0 only. x table is 16MB -> L2 hits.
    float xs = x[s * 4];
    float xd = x[d * 4];
    float ld = (xd - xs) / a;              // exact f32 divide (cheap vs memory)
    float* p = acc + d * 2;
    unsafeAtomicAdd(p,     ld);            // global_atomic_add_f32 (no return)
    unsafeAtomicAdd(p + 1, 1.0f);
}

__global__ void sdo_edge_kernel(const float* __restrict__ x,
                                const long long* __restrict__ src,
                                const long long* __restrict__ dst,
                                const float* __restrict__ attr,  // (E,2) interleaved
                                float* __restrict__ acc,
                                long long n_edges) {
    long long tid    = (long long)blockIdx.x * blockDim.x + threadIdx.x;
    long long stride = (long long)gridDim.x * blockDim.x;
    for (long long e0 = tid * 4; e0 < n_edges; e0 += stride * 4) {
        if (e0 + 4 <= n_edges) {
            // Non-temporal 128-bit streaming loads: 2 int64 indices each,
            // 4 floats (= 2 edges worth) of attr each.
            v2u64 s01 = __builtin_nontemporal_load((const v2u64*)(src + e0));
            v2u64 s23 = __builtin_nontemporal_load((const v2u64*)(src + e0 + 2));
            v2u64 d01 = __builtin_nontemporal_load((const v2u64*)(dst + e0));
            v2u64 d23 = __builtin_nontemporal_load((const v2u64*)(dst + e0 + 2));
            v4f   a01 = __builtin_nontemporal_load((const v4f*)(attr + 2 * e0));
            v4f   a23 = __builtin_nontemporal_load((const v4f*)(attr + 2 * e0 + 4));

            sdo_process_edge(x, acc, (long long)s01.x, (long long)d01.x, a01.x);
            sdo_process_edge(x, acc, (long long)s01.y, (long long)d01.y, a01.z);
            sdo_process_edge(x, acc, (long long)s23.x, (long long)d23.x, a23.x);
            sdo_process_edge(x, acc, (long long)s23.y, (long long)d23.y, a23.z);
        } else {
            for (long long e = e0; e < n_edges; ++e) {
                sdo_process_edge(x, acc, src[e], dst[e], attr[2 * e]);
            }
        }
    }
}

__global__ void sdo_finalize_kernel(const float* __restrict__ acc,
                                    float* __restrict__ out,
                                    long long n_nodes) {
    long long i = (long long)blockIdx.x * blockDim.x + threadIdx.x;
    long long stride = (long long)gridDim.x * blockDim.x;
    for (; i < n_nodes; i += stride) {
        v2f sc = *(const v2f*)(acc + 2 * i);   // (sum, count)
        out[i] = (sc.y > 0.0f) ? (sc.x / sc.y) : 0.0f;
    }
}

extern "C" void kernel_launch(void* const* d_in, const int* in_sizes, int n_in,
                              void* d_out, int out_size, void* d_ws, size_t ws_size,
                              hipStream_t stream) {
    const float*     x    = (const float*)d_in[0];      // (N,4) f32
    const long long* ei   = (const long long*)d_in[1];  // (2,E) i64
    const float*     attr = (const float*)d_in[2];      // (E,2) f32
    float*           out  = (float*)d_out;
    float*           acc  = (float*)d_ws;               // float2 per node (8MB)

    const long long n_nodes = (long long)in_sizes[0] / 4;
    const long long n_edges = (long long)in_sizes[1] / 2;
    const long long* src = ei;
    const long long* dst = ei + n_edges;

    const int tb = 256;  // 8 wave32 waves per block

    // 1) zero the (sum,count) accumulator (d_ws is poisoned by the harness)
    long long acc_floats = 2 * n_nodes;
    int zb = (int)((acc_floats / 4 + tb - 1) / tb);
    sdo_zero_kernel<<<zb, tb, 0, stream>>>(acc, acc_floats);

    // 2) edge scatter-accumulate, 4 edges per thread
    long long work = (n_edges + 3) / 4;
    int eb = (int)((work + tb - 1) / tb);
    sdo_edge_kernel<<<eb, tb, 0, stream>>>(x, src, dst, attr, acc, n_edges);

    // 3) finalize mean
    int fb = (int)((n_nodes + tb - 1) / tb);
    sdo_finalize_kernel<<<fb, tb, 0, stream>>>(acc, out, n_nodes);
}